// Attention_4398046511861
// MI455X (gfx1250) — compile-verified
//
#include <hip/hip_runtime.h>
#include <math.h>

// Problem constants (from reference)
#define S_DIM 2048
#define B_DIM 32
#define E_DIM 1024
#define A_DIM 128

typedef float v2f __attribute__((ext_vector_type(2)));
typedef float v8f __attribute__((ext_vector_type(8)));

// ---------------------------------------------------------------------------
// Generic small f32 WMMA GEMM:  C[M×N] = A[M×K] @ B (+ bias over N)
//   BT=false: B stored row-major [K×N]  (element B[k][n] = Bm[k*ldb + n])
//   BT=true : B stored row-major [N×K]  (element B[k][n] = Bm[n*ldb + k])
// One wave computes one 16×16 tile via V_WMMA_F32_16X16X4_F32 (K stepped by 4).
// A-frag (16×4 f32): lanes 0-15 -> M rows, VGPR0=K0 VGPR1=K1; lanes 16-31 -> K2,K3.
// C/D (16×16 f32): lane&15 -> N, vgpr v -> M=v (+8 for upper lane half).
// ---------------------------------------------------------------------------
template <bool BT>
__global__ void wmma_gemm_f32(const float* __restrict__ A, const float* __restrict__ Bm,
                              const float* __restrict__ bias, float* __restrict__ C,
                              int M, int N, int K, int lda, int ldb, int ldc) {
  const int lane = threadIdx.x & 31;
  const int wave = threadIdx.x >> 5;
  const int wpb  = blockDim.x >> 5;
  const int tile = blockIdx.x * wpb + wave;
  const int tilesN = N >> 4;
  const int tm = (tile / tilesN) << 4;
  const int tn = (tile % tilesN) << 4;
  if (tm >= M) return;  // wave-uniform: EXEC stays all-ones for WMMA

  const int half = lane >> 4;   // 0: K0/K1, 1: K2/K3
  const int l15  = lane & 15;

  v8f acc = {};
  const float* arow = A + (size_t)(tm + l15) * lda;
  for (int k = 0; k < K; k += 4) {
    const int kb = k + (half << 1);
    v2f a, b;
    a.x = arow[kb];
    a.y = arow[kb + 1];
    if (BT) {
      const float* bcol = Bm + (size_t)(tn + l15) * ldb;
      b.x = bcol[kb];
      b.y = bcol[kb + 1];
    } else {
      b.x = Bm[(size_t)kb * ldb + (tn + l15)];
      b.y = Bm[(size_t)(kb + 1) * ldb + (tn + l15)];
    }
    // 8 args: (neg_a, A, neg_b, B, c_mod, C, reuse_a, reuse_b)
    acc = __builtin_amdgcn_wmma_f32_16x16x4_f32(false, a, false, b, (short)0, acc,
                                                false, false);
  }

  const float bv = bias ? bias[tn + l15] : 0.0f;
#pragma unroll
  for (int vv = 0; vv < 8; ++vv) {
    const int row = tm + vv + (half << 3);
    C[(size_t)row * ldc + (tn + l15)] = acc[vv] + bv;
  }
}

// c[b] = dot(b_ae, proj_d[b,:])   (tiny: 32 threads x 128 MACs)
__global__ void compute_c_kernel(const float* __restrict__ proj_d,
                                 const float* __restrict__ b_ae,
                                 float* __restrict__ cvec) {
  const int b = threadIdx.x;
  if (b < B_DIM) {
    float s = 0.0f;
#pragma unroll 8
    for (int a = 0; a < A_DIM; ++a) s += b_ae[a] * proj_d[b * A_DIM + a];
    cvec[b] = s;
  }
}

// scores[s,b] = dot(enc[s,b,:], v[b,:]) + c[b]  — one wave per (s,b) row.
// float4 (global_load_b128) fully coalesced streaming of the 256 MB enc tensor.
__global__ void scores_kernel(const float* __restrict__ enc, const float* __restrict__ v,
                              const float* __restrict__ cvec, float* __restrict__ scores) {
  const int wid  = (int)((blockIdx.x * blockDim.x + threadIdx.x) >> 5);
  const int lane = threadIdx.x & 31;
  const int s = wid >> 5;       // / B_DIM
  const int b = wid & (B_DIM - 1);

  const float4* erow = (const float4*)(enc + ((size_t)s * B_DIM + b) * E_DIM);
  const float4* vrow = (const float4*)(v + (size_t)b * E_DIM);

  float acc = 0.0f;
#pragma unroll
  for (int i = 0; i < E_DIM / 128; ++i) {   // 8 iterations of 128-float strips
    const float4 e4 = erow[i * 32 + lane];
    const float4 v4 = vrow[i * 32 + lane];
    acc += e4.x * v4.x + e4.y * v4.y + e4.z * v4.z + e4.w * v4.w;
  }
#pragma unroll
  for (int off = 16; off; off >>= 1) acc += __shfl_xor(acc, off, 32);
  if (lane == 0) scores[s * B_DIM + b] = acc + cvec[b];
}

// In-place softmax over the sequence dim: one block (256 thr) per batch b.
__global__ void softmax_kernel(float* __restrict__ scores) {
  const int b   = blockIdx.x;
  const int tid = threadIdx.x;
  __shared__ float red[8];

  float vals[S_DIM / 256];
  float m = -3.402823466e38f;
#pragma unroll
  for (int i = 0; i < S_DIM / 256; ++i) {
    vals[i] = scores[(i * 256 + tid) * B_DIM + b];
    m = fmaxf(m, vals[i]);
  }
#pragma unroll
  for (int off = 16; off; off >>= 1) m = fmaxf(m, __shfl_xor(m, off, 32));
  if ((tid & 31) == 0) red[tid >> 5] = m;
  __syncthreads();
  float bm = red[0];
#pragma unroll
  for (int i = 1; i < 8; ++i) bm = fmaxf(bm, red[i]);
  __syncthreads();

  float sum = 0.0f;
#pragma unroll
  for (int i = 0; i < S_DIM / 256; ++i) {
    vals[i] = __expf(vals[i] - bm);
    sum += vals[i];
  }
#pragma unroll
  for (int off = 16; off; off >>= 1) sum += __shfl_xor(sum, off, 32);
  if ((tid & 31) == 0) red[tid >> 5] = sum;
  __syncthreads();
  float bs = 0.0f;
#pragma unroll
  for (int i = 0; i < 8; ++i) bs += red[i];
  const float inv = 1.0f / bs;
#pragma unroll
  for (int i = 0; i < S_DIM / 256; ++i)
    scores[(i * 256 + tid) * B_DIM + b] = vals[i] * inv;   // in-place -> alphas
}

// context partials: grid = B_DIM * (E_DIM/256) * 8 s-chunks = 1024 blocks.
// Each block streams 256 s-steps of a 256-float e-strip (coalesced 1 KB / iter).
__global__ void context_partial_kernel(const float* __restrict__ enc,
                                       const float* __restrict__ alphas,
                                       float* __restrict__ partial) {
  const int blk = blockIdx.x;
  const int sc  = blk & 7;           // s-chunk
  const int ec  = (blk >> 3) & 3;    // e-chunk
  const int b   = blk >> 5;          // batch
  const int e   = ec * 256 + threadIdx.x;
  const int s0  = sc * (S_DIM / 8);

  float acc = 0.0f;
#pragma unroll 4
  for (int i = 0; i < S_DIM / 8; ++i) {
    const int s = s0 + i;
    acc += alphas[s * B_DIM + b] * enc[((size_t)s * B_DIM + b) * E_DIM + e];
  }
  partial[((size_t)sc * B_DIM + b) * E_DIM + e] = acc;
}

// Deterministic final reduce of the 8 s-chunk partials -> context[b,e].
__global__ void context_reduce_kernel(const float* __restrict__ partial,
                                      float* __restrict__ out) {
  const int idx = blockIdx.x * blockDim.x + threadIdx.x;  // 0..32767
  float acc = 0.0f;
#pragma unroll
  for (int sc = 0; sc < 8; ++sc)
    acc += partial[(size_t)sc * (B_DIM * E_DIM) + idx];
  out[idx] = acc;
}

extern "C" void kernel_launch(void* const* d_in, const int* in_sizes, int n_in,
                              void* d_out, int out_size, void* d_ws, size_t ws_size,
                              hipStream_t stream) {
  (void)in_sizes; (void)n_in; (void)out_size; (void)ws_size;

  const float* enc  = (const float*)d_in[0];  // [S,B,E]
  const float* dec  = (const float*)d_in[1];  // [B,D]
  const float* w_ae = (const float*)d_in[2];  // [A,E]
  const float* b_ae = (const float*)d_in[3];  // [A]
  const float* w_ad = (const float*)d_in[4];  // [A,D]
  const float* b_ad = (const float*)d_in[5];  // [A]
  float* out = (float*)d_out;                 // [B,E]

  // Workspace carving (floats), ~1.5 MB total
  float* proj_d  = (float*)d_ws;              // B*A      = 4096
  float* vmat    = proj_d + (B_DIM * A_DIM);  // B*E      = 32768
  float* cvec    = vmat + (B_DIM * E_DIM);    // 64 (padded)
  float* scores  = cvec + 64;                 // S*B      = 65536 (alphas in-place)
  float* partial = scores + (S_DIM * B_DIM);  // 8*B*E    = 262144

  // 1) proj_d[b,a] = dec[b,:] @ w_ad[a,:] + b_ad   (M=32,N=128,K=1024; B transposed)
  wmma_gemm_f32<true><<<2, 256, 0, stream>>>(dec, w_ad, b_ad, proj_d,
                                             B_DIM, A_DIM, E_DIM, E_DIM, E_DIM, A_DIM);
  // 2) v[b,e] = proj_d[b,:] @ w_ae[:,e]            (M=32,N=1024,K=128; B row-major K×N)
  wmma_gemm_f32<false><<<16, 256, 0, stream>>>(proj_d, w_ae, nullptr, vmat,
                                               B_DIM, E_DIM, A_DIM, A_DIM, E_DIM, E_DIM);
  // 2b) c[b] = dot(b_ae, proj_d[b,:])
  compute_c_kernel<<<1, 32, 0, stream>>>(proj_d, b_ae, cvec);
  // 3) scores: streaming pass 1 over enc (one wave per (s,b) row)
  scores_kernel<<<(S_DIM * B_DIM) / 8, 256, 0, stream>>>(enc, vmat, cvec, scores);
  // 4) softmax over s (in-place -> alphas)
  softmax_kernel<<<B_DIM, 256, 0, stream>>>(scores);
  // 5) context partials: streaming pass 2 over enc
  context_partial_kernel<<<B_DIM * 4 * 8, 256, 0, stream>>>(enc, scores, partial);
  // 6) deterministic reduce -> out[b,e]
  context_reduce_kernel<<<(B_DIM * E_DIM) / 256, 256, 0, stream>>>(partial, out);
}